// SeparableConvModule_47253230191041
// MI455X (gfx1250) — compile-verified
//
#include <hip/hip_runtime.h>

typedef float v2f __attribute__((ext_vector_type(2)));
typedef float v8f __attribute__((ext_vector_type(8)));

namespace {
constexpr int NB = 2, NC = 3, NH = 512, NW = 512, NFS = 51;
constexpr int HPimg = NH + NFS - 1;   // 562
constexpr int WPimg = NW + NFS - 1;   // 562
constexpr int RPitch   = 116;  // staged input-row pitch (>=64+50, even->aligned b64, 116%64=52 -> conflict-free)
constexpr int HBandPad = 16;   // j offset in banded kernel tile (j in [-15,67])
constexpr int HPitch   = 84;   // banded entries per x
constexpr int VPitch   = 65;
constexpr int KT       = 17;   // ceil(66/4) K-steps of WMMA K=4
}

__global__ __launch_bounds__(128) void sepconv_wmma_f32(
    const float* __restrict__ img,    // [B, C, 562, 562]
    const float* __restrict__ vker,   // [B, 51, 512, 512] vertical
    const float* __restrict__ hker,   // [B, 51, 512, 512] horizontal
    float* __restrict__ out)          // [B, C, 512, 512]
{
  __shared__ float Rl[64 * RPitch];   // input rows, current channel: Rl[i][t]
  __shared__ float Hl[64 * HPitch];   // banded horiz kernels: Hl[x][HBandPad + j], zero outside band
  __shared__ float Vl[NFS * VPitch];  // vertical kernels: Vl[i][x]

  const int tid  = threadIdx.x;
  const int wave = tid >> 5;
  const int lane = tid & 31;
  const int xl   = lane & 15;
  const int half = lane >> 4;
  const int y    = blockIdx.y;
  const int b    = blockIdx.z;
  const int x0   = blockIdx.x * 64;
  const int xb   = wave * 16 + xl;    // this lane's x within the block

  // Stage banded horizontal kernels: Hl[x][16 + j] = h[j][x], zeros outside [0,51)
  for (int idx = tid; idx < 64 * HPitch; idx += 128) {
    const int x = idx / HPitch;
    const int j = idx % HPitch - HBandPad;
    float v = 0.f;
    if (j >= 0 && j < NFS)
      v = hker[(((size_t)b * NFS + j) * NH + y) * NW + (x0 + x)];
    Hl[idx] = v;
  }
  // Stage vertical kernels
  for (int idx = tid; idx < NFS * 64; idx += 128) {
    const int i = idx >> 6;
    const int x = idx & 63;
    Vl[i * VPitch + x] = vker[(((size_t)b * NFS + i) * NH + y) * NW + (x0 + x)];
  }

  for (int c = 0; c < NC; ++c) {
    __syncthreads();   // previous channel's Rl reads complete
    // Stage 51 input rows for this channel; fill full pitch (clamped; dead lanes hit zero band)
    for (int idx = tid; idx < NFS * RPitch; idx += 128) {
      const int i = idx / RPitch;
      int gx = x0 + idx % RPitch;
      gx = gx < WPimg ? gx : (WPimg - 1);
      Rl[idx] = img[(((size_t)b * NC + c) * HPimg + (y + i)) * WPimg + gx];
    }
    __syncthreads();

    v8f acc[4];
    #pragma unroll
    for (int mt = 0; mt < 4; ++mt) {
      v8f z = {0.f, 0.f, 0.f, 0.f, 0.f, 0.f, 0.f, 0.f};
      acc[mt] = z;
    }

    // Horizontal stage: D[row=i, n=x] += R[row, t] * band(h)[t, x]
    for (int kt = 0; kt < KT; ++kt) {
      const int t0 = kt * 4 + half * 2;          // K pair handled by this lane half
      const int hb = xb * HPitch + HBandPad + t0 - xl;
      v2f bm;                                     // B: band matrix, N=lane
      bm.x = Hl[hb];
      bm.y = Hl[hb + 1];
      #pragma unroll
      for (int mt = 0; mt < 4; ++mt) {
        const int ab = (mt * 16 + xl) * RPitch + wave * 16 + t0;  // even -> 8B aligned
        v2f am = *(const v2f*)&Rl[ab];            // A: rows of R, M=lane
        acc[mt] = __builtin_amdgcn_wmma_f32_16x16x4_f32(
            false, am, false, bm, (short)0, acc[mt], false, false);
      }
    }

    // Vertical stage: out[c][x] = sum_i v[i][x] * HF[i][x]; lane holds fixed x
    float o = 0.f;
    #pragma unroll
    for (int mt = 0; mt < 4; ++mt) {
      #pragma unroll
      for (int r = 0; r < 8; ++r) {
        const int row = mt * 16 + half * 8 + r;   // D layout: vgpr r -> row, +8 for upper half
        if (row < NFS)
          o += acc[mt][r] * Vl[row * VPitch + xb];
      }
    }
    o += __shfl_xor(o, 16, 32);                   // combine the two lane halves (same x)
    if (lane < 16)
      out[(((size_t)b * NC + c) * NH + y) * NW + (x0 + xb)] = o;
  }
}

extern "C" void kernel_launch(void* const* d_in, const int* in_sizes, int n_in,
                              void* d_out, int out_size, void* d_ws, size_t ws_size,
                              hipStream_t stream) {
  (void)in_sizes; (void)n_in; (void)out_size; (void)d_ws; (void)ws_size;
  const float* img  = (const float*)d_in[0];   // input1
  const float* vker = (const float*)d_in[1];   // input2 (vertical)
  const float* hker = (const float*)d_in[2];   // input3 (horizontal)
  float* out = (float*)d_out;
  dim3 grid(NW / 64, NH, NB);
  sepconv_wmma_f32<<<grid, dim3(128), 0, stream>>>(img, vker, hker, out);
}